// NG_21457656611332
// MI455X (gfx1250) — compile-verified
//
#include <hip/hip_runtime.h>

typedef float v2f __attribute__((ext_vector_type(2)));
typedef float v8f __attribute__((ext_vector_type(8)));

#define NODES 8192
#define IN_F 1024
#define OUT_F 1024
#define BATCH 32
#define WSTRIDE (NODES - 1) /* 8191 floats per weights row */

__device__ __forceinline__ v2f ld2(const float* p) {
  // weights rows have odd element stride (8191) -> only 4B alignment
  // guaranteed; load scalars and let the backend combine when legal.
  v2f r;
  r.x = p[0];
  r.y = p[1];
  return r;
}

// One block per 16-column output tile (64 blocks). 256 threads = 8 waves;
// each wave owns a K-chunk of 128 (32 WMMA k-steps) and accumulates a
// 32x16 f32 tile as two 16x16 WMMA accumulators. LDS tree-reduce at the end.
__global__ __launch_bounds__(256) void NG_gemm_f32_wmma(
    const float* __restrict__ x,     // [32, 1024]
    const float* __restrict__ w,     // [8192, 8191]
    const float* __restrict__ bias,  // [8192]
    float* __restrict__ out) {       // [32, 1024]
  const int tid  = threadIdx.x;
  const int lane = tid & 31;
  const int wave = tid >> 5;   // 0..7
  const int l16  = lane & 15;  // M index (A) / N index (B,C,D)
  const int h    = lane >> 4;  // selects K pair {0,1} vs {2,3}
  const int nt   = blockIdx.x; // 0..63 output column tile

  // Output column (nt*16 + l16) reads weights row (8191 - that column).
  const long wrow  = (long)(NODES - 1 - (nt * 16 + l16)) * WSTRIDE;
  const int  kbase = wave * 128 + 2 * h;

  const float* wp  = w + wrow + kbase;
  const float* xp0 = x + l16 * IN_F + kbase; // batch rows 0..15
  const float* xp1 = xp0 + 16 * IN_F;        // batch rows 16..31

  v8f c0 = {0.f, 0.f, 0.f, 0.f, 0.f, 0.f, 0.f, 0.f};
  v8f c1 = c0;

#pragma unroll 8
  for (int ks = 0; ks < 32; ++ks) {
    v2f a0 = ld2(xp0 + ks * 4); // A tile rows 0..15,  K = kbase+4ks .. +1
    v2f a1 = ld2(xp1 + ks * 4); // A tile rows 16..31
    v2f b  = ld2(wp + ks * 4);  // shared B fragment
    // D = A * B + C, full f32 (V_WMMA_F32_16X16X4_F32)
    c0 = __builtin_amdgcn_wmma_f32_16x16x4_f32(false, a0, false, b,
                                               (short)0, c0, false, false);
    c1 = __builtin_amdgcn_wmma_f32_16x16x4_f32(false, a1, false, b,
                                               (short)0, c1, false, false);
  }

  // Cross-wave reduction of the 8 partial 32x16 tiles.
  __shared__ float red[8 * 512];
  float* my = red + wave * 512;
#pragma unroll
  for (int r = 0; r < 8; ++r) {
    const int m = r + 8 * h; // C/D layout: VGPR r -> M=r (lanes 0-15), M=r+8
    my[m * 16 + l16]        = c0[r];
    my[(16 + m) * 16 + l16] = c1[r];
  }
  __syncthreads();

#pragma unroll
  for (int i = 0; i < 2; ++i) {
    const int e = tid + 256 * i; // 0..511 over the 32x16 tile
    float s = 0.f;
#pragma unroll
    for (int wv = 0; wv < 8; ++wv) s += red[wv * 512 + e];
    const int m = e >> 4;
    const int n = e & 15;
    const int kout = nt * 16 + n;
    out[m * OUT_F + kout] = s + bias[NODES - 1 - kout];
  }
}

extern "C" void kernel_launch(void* const* d_in, const int* in_sizes, int n_in,
                              void* d_out, int out_size, void* d_ws,
                              size_t ws_size, hipStream_t stream) {
  const float* x    = (const float*)d_in[0]; // [32,1024]
  const float* w    = (const float*)d_in[1]; // [8192,8191]
  const float* bias = (const float*)d_in[2]; // [8192]
  float* out        = (float*)d_out;         // [32,1024]
  (void)in_sizes; (void)n_in; (void)out_size; (void)d_ws; (void)ws_size;

  NG_gemm_f32_wmma<<<dim3(OUT_F / 16), dim3(256), 0, stream>>>(x, w, bias, out);
}